// GATForHetero_78391743087102
// MI455X (gfx1250) — compile-verified
//
#include <hip/hip_runtime.h>
#include <math.h>

// ---------------- problem constants (from reference) ----------------
constexpr int Df  = 128;   // input feature dim
constexpr int Hh  = 4;     // heads
constexpr int Cc  = 32;    // channels per head
constexpr int HC  = 128;   // Hh*Cc
constexpr int LL  = 3;     // layers
constexpr float NEG_SLOPE = 0.2f;
constexpr float EPSF      = 1e-16f;

typedef __attribute__((ext_vector_type(2))) float v2f;
typedef __attribute__((ext_vector_type(8))) float v8f;

// ---------------- small utility kernels ----------------
__global__ void gat_fill(float* __restrict__ p, float v, int n) {
    int i = blockIdx.x * blockDim.x + threadIdx.x;
    if (i < n) p[i] = v;
}

// out[n*128 + k] = bias[k]  (segment_sum accumulates on top -> sum + bias)
__global__ void gat_init_out(float* __restrict__ out, const float* __restrict__ bias, int total) {
    int i = blockIdx.x * blockDim.x + threadIdx.x;
    if (i < total) out[i] = bias[i & (HC - 1)];
}

__global__ void gat_relu(float* __restrict__ p, int n) {
    int i = blockIdx.x * blockDim.x + threadIdx.x;
    if (i < n) { float v = p[i]; p[i] = v > 0.f ? v : 0.f; }
}

// ---------------- weight repack into WMMA B-fragment order ----------------
// Wt[((coltile*32 + kstep)*32 + lane)] = { W[ka][col], W[ka+1][col] }
//   ka  = kstep*4 + 2*(lane>>4),  col = coltile*16 + (lane&15)
// -> in the GEMM, each wave's per-kstep B fragment is 32 consecutive float2
//    (256 B contiguous, one coalesced global_load_b64 per lane).
__global__ void gat_pack_w(const float* __restrict__ Wsl, const float* __restrict__ Wdl,
                           float2* __restrict__ WtS, float2* __restrict__ WtD)
{
    int i = blockIdx.x * blockDim.x + threadIdx.x;      // 0 .. 2*8192-1
    const int which = i >> 13;                          // 0: Ws, 1: Wd
    const int j     = i & 8191;
    const float*  W  = which ? Wdl : Wsl;
    float2*       Wt = which ? WtD : WtS;
    const int lane    = j & 31;
    const int kstep   = (j >> 5) & 31;
    const int coltile = j >> 10;                        // 0..7
    const int ka  = kstep * 4 + 2 * (lane >> 4);
    const int col = coltile * 16 + (lane & 15);
    float2 v;
    v.x = W[(size_t)ka       * HC + col];
    v.y = W[(size_t)(ka + 1) * HC + col];
    Wt[j] = v;
}

// ---------------- fp32 WMMA GEMM: xs = h @ W  (N x 128 @ 128 x 128) ----------------
// One wave computes a 16x16 output tile via 32 chained V_WMMA_F32_16X16X4_F32.
// Block = 256 threads = 8 waves -> 16 rows x 128 cols per block.
// grid.y == 0 -> (WtS -> xs), grid.y == 1 -> (WtD -> xd).
__global__ __launch_bounds__(256)
void gat_gemm_wmma(const float* __restrict__ h,
                   const float2* __restrict__ WtS,  // repacked B fragments
                   const float2* __restrict__ WtD,
                   float* __restrict__ xs,
                   float* __restrict__ xd,
                   int N)
{
    __shared__ float lds[16][132];   // 16x128 A-tile, padded stride to avoid bank conflicts

    const int tid  = threadIdx.x;
    const int wave = tid >> 5;       // 0..7 -> 16-col chunk
    const int lane = tid & 31;
    const int row0 = blockIdx.x * 16;

    const float2* __restrict__ Wt  = (blockIdx.y == 0) ? WtS : WtD;
    float*        __restrict__ out = (blockIdx.y == 0) ? xs  : xd;

    // cooperative load: 256 threads x 8 floats = 16x128 tile (N divisible by 16)
    {
        const int r  = tid >> 4;          // 0..15
        const int c0 = (tid & 15) * 8;    // 0..120 step 8
        const float4* src = reinterpret_cast<const float4*>(h + (size_t)(row0 + r) * Df + c0);
        float4 v0 = src[0];
        float4 v1 = src[1];
        lds[r][c0 + 0] = v0.x; lds[r][c0 + 1] = v0.y; lds[r][c0 + 2] = v0.z; lds[r][c0 + 3] = v0.w;
        lds[r][c0 + 4] = v1.x; lds[r][c0 + 5] = v1.y; lds[r][c0 + 6] = v1.z; lds[r][c0 + 7] = v1.w;
    }
    __syncthreads();

    const int half = lane >> 4;      // 0: lanes 0-15, 1: lanes 16-31
    const int l16  = lane & 15;

    // this wave's B fragment stream: 32 steps x 32 lanes of float2
    const float2* __restrict__ wp = Wt + ((size_t)wave * 32 * 32 + lane);

    v8f acc = {};
    #pragma unroll
    for (int ks = 0; ks < 32; ++ks) {
        const int ka = ks * 4 + 2 * half;
        // A fragment (16x4): lane holds row M=l16; VGPR v holds K = ks*4 + 2*half + v
        v2f a;
        a.x = lds[l16][ka];
        a.y = lds[l16][ka + 1];
        // B fragment (4x16): prepacked, coalesced 8-byte load
        float2 bw = wp[(size_t)ks * 32];
        v2f b;
        b.x = bw.x;
        b.y = bw.y;
        acc = __builtin_amdgcn_wmma_f32_16x16x4_f32(false, a, false, b,
                                                    (short)0, acc, false, false);
    }

    // D tile: VGPR j holds row M = j + 8*half, col N = l16
    float* op = out + (size_t)row0 * HC + wave * 16;
    #pragma unroll
    for (int j = 0; j < 8; ++j) {
        const int m = j + 8 * half;
        op[(size_t)m * HC + l16] = acc[j];
    }
}

// ---------------- per-(node,head) attention logits ----------------
__global__ void gat_alpha(const float* __restrict__ xs, const float* __restrict__ xd,
                          const float* __restrict__ as_l,   // [H,C] slice
                          const float* __restrict__ ad_l,
                          float* __restrict__ alpha_s, float* __restrict__ alpha_d,
                          int NH)
{
    int i = blockIdx.x * blockDim.x + threadIdx.x;
    if (i >= NH) return;
    const int n = i >> 2;
    const int h = i & 3;
    const float4* xsv = reinterpret_cast<const float4*>(xs + (size_t)n * HC + h * Cc);
    const float4* xdv = reinterpret_cast<const float4*>(xd + (size_t)n * HC + h * Cc);
    const float4* av  = reinterpret_cast<const float4*>(as_l + h * Cc);
    const float4* bv  = reinterpret_cast<const float4*>(ad_l + h * Cc);
    float ss = 0.f, sd = 0.f;
    #pragma unroll
    for (int j = 0; j < 8; ++j) {
        float4 a = xsv[j], w = av[j];
        ss += a.x * w.x + a.y * w.y + a.z * w.z + a.w * w.w;
        float4 d = xdv[j], u = bv[j];
        sd += d.x * u.x + d.y * u.y + d.z * u.z + d.w * u.w;
    }
    alpha_s[i] = ss;
    alpha_d[i] = sd;
}

// ---------------- edge phase ----------------
__device__ __forceinline__ void atomicMaxF32(float* addr, float v) {
    if (v >= 0.f) atomicMax(reinterpret_cast<int*>(addr), __float_as_int(v));
    else          atomicMin(reinterpret_cast<unsigned int*>(addr), __float_as_uint(v));
}

// e = leaky_relu(alpha_s[src] + alpha_d[dst]); segment max over dst
__global__ void gat_edge_logits(const int* __restrict__ src, const int* __restrict__ dst,
                                const float* __restrict__ alpha_s, const float* __restrict__ alpha_d,
                                float* __restrict__ ebuf, float* __restrict__ emax, int EH)
{
    int i = blockIdx.x * blockDim.x + threadIdx.x;
    if (i >= EH) return;
    const int e = i >> 2;
    const int h = i & 3;
    const int s = src[e];
    const int d = dst[e];
    float v = alpha_s[(size_t)s * Hh + h] + alpha_d[(size_t)d * Hh + h];
    v = v > 0.f ? v : NEG_SLOPE * v;
    ebuf[i] = v;
    atomicMaxF32(emax + (size_t)d * Hh + h, v);
}

// ex = exp(e - emax[dst]); segment sum over dst
__global__ void gat_edge_exp(const int* __restrict__ dst,
                             float* __restrict__ ebuf, const float* __restrict__ emax,
                             float* __restrict__ denom, int EH)
{
    int i = blockIdx.x * blockDim.x + threadIdx.x;
    if (i >= EH) return;
    const int e = i >> 2;
    const int h = i & 3;
    const int d = dst[e];
    float ex = __expf(ebuf[i] - emax[(size_t)d * Hh + h]);
    ebuf[i] = ex;
    atomicAdd(denom + (size_t)d * Hh + h, ex);
}

// out[dst] += (ex/denom[dst]) * xs[src] ; one wave per edge, float4 per lane
__global__ __launch_bounds__(256)
void gat_scatter(const int* __restrict__ src, const int* __restrict__ dst,
                 const float* __restrict__ ex, const float* __restrict__ denom,
                 const float* __restrict__ xs, float* __restrict__ out, int E)
{
    const int edge = (blockIdx.x * blockDim.x + threadIdx.x) >> 5;
    const int lane = threadIdx.x & 31;
    if (edge >= E) return;
    const int s = src[edge];
    const int d = dst[edge];
    const int head = lane >> 3;                 // 4 floats per lane, 8 lanes per head
    const float w = ex[(size_t)edge * Hh + head] /
                    (denom[(size_t)d * Hh + head] + EPSF);
    const float4* xv = reinterpret_cast<const float4*>(xs + (size_t)s * HC);
    float4 v = xv[lane];
    float* op = out + (size_t)d * HC + lane * 4;
    atomicAdd(op + 0, v.x * w);
    atomicAdd(op + 1, v.y * w);
    atomicAdd(op + 2, v.z * w);
    atomicAdd(op + 3, v.w * w);
}

// ---------------- host-side orchestration ----------------
extern "C" void kernel_launch(void* const* d_in, const int* in_sizes, int n_in,
                              void* d_out, int out_size, void* d_ws, size_t ws_size,
                              hipStream_t stream)
{
    (void)n_in; (void)out_size; (void)ws_size;
    const float* x   = (const float*)d_in[0];
    const int*   ei  = (const int*)  d_in[1];
    const float* Ws  = (const float*)d_in[2];
    const float* Wd  = (const float*)d_in[3];
    const float* as_ = (const float*)d_in[4];
    const float* ad_ = (const float*)d_in[5];
    const float* bb  = (const float*)d_in[6];

    const int N = in_sizes[0] / HC;     // 100000 (divisible by 16)
    const int E = in_sizes[1] / 2;      // 1600000
    const int* srcI = ei;
    const int* dstI = ei + E;

    // workspace layout (floats)
    float* hbuf    = (float*)d_ws;                      // N*128
    float* xs      = hbuf    + (size_t)N * HC;          // N*128
    float* xd      = xs      + (size_t)N * HC;          // N*128
    float* ebuf    = xd      + (size_t)N * HC;          // E*H
    float* alpha_s = ebuf    + (size_t)E * Hh;          // N*H
    float* alpha_d = alpha_s + (size_t)N * Hh;          // N*H
    float* emax    = alpha_d + (size_t)N * Hh;          // N*H
    float* denom   = emax    + (size_t)N * Hh;          // N*H
    float2* WtS    = (float2*)(denom + (size_t)N * Hh); // 8192 float2 (64 KB)
    float2* WtD    = WtS + 8192;                        // 8192 float2 (64 KB)

    const int NH = N * Hh;
    const int EH = E * Hh;

    for (int l = 0; l < LL; ++l) {
        const float* hin  = (l == 0)      ? x            : hbuf;
        float*       hout = (l == LL - 1) ? (float*)d_out : hbuf;
        const float* Wsl  = Ws  + (size_t)l * Df * HC;
        const float* Wdl  = Wd  + (size_t)l * Df * HC;
        const float* asl  = as_ + (size_t)l * Hh * Cc;
        const float* adl  = ad_ + (size_t)l * Hh * Cc;
        const float* bl   = bb  + (size_t)l * HC;

        // 0) repack this layer's weights into B-fragment order (2*8192 float2)
        gat_pack_w<<<(2 * 8192) / 256, 256, 0, stream>>>(Wsl, Wdl, WtS, WtD);

        // 1) projections via fp32 WMMA
        dim3 gg(N / 16, 2);
        gat_gemm_wmma<<<gg, 256, 0, stream>>>(hin, WtS, WtD, xs, xd, N);

        // 2) per-(node,head) logits
        gat_alpha<<<(NH + 255) / 256, 256, 0, stream>>>(xs, xd, asl, adl, alpha_s, alpha_d, NH);

        // 3) reset segment accumulators; preload bias into output accumulator
        gat_fill<<<(NH + 255) / 256, 256, 0, stream>>>(emax,  -INFINITY, NH);
        gat_fill<<<(NH + 255) / 256, 256, 0, stream>>>(denom, 0.f,       NH);
        gat_init_out<<<(N * HC + 255) / 256, 256, 0, stream>>>(hout, bl, N * HC);

        // 4) segment softmax over destination neighborhoods
        gat_edge_logits<<<(EH + 255) / 256, 256, 0, stream>>>(srcI, dstI, alpha_s, alpha_d,
                                                              ebuf, emax, EH);
        gat_edge_exp<<<(EH + 255) / 256, 256, 0, stream>>>(dstI, ebuf, emax, denom, EH);

        // 5) weighted scatter-aggregate (one wave per edge)
        gat_scatter<<<(E + 7) / 8, 256, 0, stream>>>(srcI, dstI, ebuf, denom, xs, hout, E);

        // 6) inter-layer ReLU
        if (l < LL - 1)
            gat_relu<<<(N * HC + 255) / 256, 256, 0, stream>>>(hout, N * HC);
    }
}